// Corr_16492674417478
// MI455X (gfx1250) — compile-verified
//
#include <hip/hip_runtime.h>

typedef __attribute__((ext_vector_type(2))) float v2f;
typedef __attribute__((ext_vector_type(8))) float v8f;

// ---- fixed problem sizes (from setup_inputs) ----
constexpr int B_ = 2, TP = 4, TF = 2;
constexpr int C0 = 128, H0 = 96, W0 = 160, HW0 = H0 * W0;   // level 0
constexpr int C1 = 256, H1 = 48, W1 = 80,  HW1 = H1 * W1;   // level 1
constexpr int C2 = 512, H2 = 24, W2 = 40,  HW2 = H2 * W2;   // level 2
constexpr int NPAIR = B_ * (TP - 1);   // 6 frame pairs
constexpr int NDISP = 81;              // 9x9 displacements

// ---------------------------------------------------------------------------
// 1) inverse L2 norm over channels for every (b,t,pixel) of features0
// ---------------------------------------------------------------------------
__global__ void norm_kernel(const float* __restrict__ f0, float* __restrict__ inorm) {
  const int p = blockIdx.x * blockDim.x + threadIdx.x;
  const int bt = blockIdx.y;
  if (p >= HW0) return;
  const float* base = f0 + (size_t)bt * C0 * HW0 + p;
  float s = 0.f;
  for (int c = 0; c < C0; ++c) { const float v = base[(size_t)c * HW0]; s += v * v; }
  inorm[(size_t)bt * HW0 + p] = 1.f / fmaxf(sqrtf(s), 1e-12f);
}

// ---------------------------------------------------------------------------
// 2) correlation dot maps via fp32 WMMA.
//    One wave per (16-pixel x-tile, row y, pair). For each of 9 dy values,
//    D = A(16px x 128ch) * B(128ch x 24 shifted cols) via 32x V_WMMA_F32_16X16X4_F32
//    per N-tile; band-extract the 9 dx values through LDS and scale by inorms.
//    Loads are unconditional from clamped addresses + cndmask selects so the
//    inner loop carries no exec-mask branches around the WMMAs.
//    dm layout: [pair][81][H0*W0]
// ---------------------------------------------------------------------------
__global__ void __launch_bounds__(32)
corr_dot_wmma_kernel(const float* __restrict__ f0,
                     const float* __restrict__ inorm,
                     float* __restrict__ dm) {
  const int lane = threadIdx.x;
  const int half = lane >> 4;      // 0: K0/K1 lanes, 1: K2/K3 lanes
  const int m    = lane & 15;      // M row (pixel) / N col index
  const int xbase = blockIdx.x * 16;
  const int y     = blockIdx.y;
  const int pb    = blockIdx.z;
  const int b = pb / (TP - 1), t = pb % (TP - 1);
  const int bt1 = b * TP + t, bt2 = bt1 + 1;
  const float* x1 = f0 + (size_t)bt1 * C0 * HW0;
  const float* x2 = f0 + (size_t)bt2 * C0 * HW0;

  // column indices for the two N-tiles, clamped so loads are always in-bounds
  const int q0 = xbase - 4 + m;              // N-tile 0: dx window left part
  const int q1 = xbase + 12 + m;             // N-tile 1: right part (8 cols used)
  const bool ok0 = (q0 >= 0) && (q0 < W0);
  const bool ok1 = (m < 8) && (q1 < W0);
  const int q0c = min(max(q0, 0), W0 - 1);
  const int q1c = min(q1, W0 - 1);

  __shared__ float Dt[16][33];

  for (int dyi = 0; dyi < 9; ++dyi) {
    const int yy = y + dyi - 4;
    const bool rowOK = (yy >= 0) && (yy < H0);     // uniform across the wave
    v8f acc0 = {0.f,0.f,0.f,0.f,0.f,0.f,0.f,0.f};
    v8f acc1 = {0.f,0.f,0.f,0.f,0.f,0.f,0.f,0.f};
    if (rowOK) {
      // strength-reduced channel walk: advance by 4*HW0 per K-step
      const float* pa  = x1 + (size_t)(2 * half) * HW0 + (size_t)y  * W0 + xbase + m;
      const float* pb0 = x2 + (size_t)(2 * half) * HW0 + (size_t)yy * W0 + q0c;
      const float* pb1 = x2 + (size_t)(2 * half) * HW0 + (size_t)yy * W0 + q1c;
      for (int kk = 0; kk < C0 / 4; ++kk) {
        v2f a;   a.x = pa[0];  a.y = pa[HW0];
        const float u0 = pb0[0], u1 = pb0[HW0];
        const float v0 = pb1[0], v1 = pb1[HW0];
        v2f b0v; b0v.x = ok0 ? u0 : 0.f; b0v.y = ok0 ? u1 : 0.f;
        v2f b1v; b1v.x = ok1 ? v0 : 0.f; b1v.y = ok1 ? v1 : 0.f;
        acc0 = __builtin_amdgcn_wmma_f32_16x16x4_f32(false, a, false, b0v, (short)0, acc0, false, false);
        acc1 = __builtin_amdgcn_wmma_f32_16x16x4_f32(false, a, false, b1v, (short)0, acc1, false, false);
        pa  += 4 * HW0;
        pb0 += 4 * HW0;
        pb1 += 4 * HW0;
      }
    }
    __syncthreads();
    #pragma unroll
    for (int r = 0; r < 8; ++r) {
      Dt[r + 8 * half][m]      = acc0[r];
      Dt[r + 8 * half][16 + m] = acc1[r];
    }
    __syncthreads();
    // band extraction: dm[(dy,dx)][y][xbase+px] = D[px][px+dx+4] * inorm1 * inorm2
    for (int i = lane; i < 144; i += 32) {
      const int px = i / 9, dxi = i % 9;
      const int q = xbase + px + dxi - 4;
      float v = 0.f;
      if (rowOK && q >= 0 && q < W0) {
        v = Dt[px][px + dxi]
          * inorm[(size_t)bt1 * HW0 + (size_t)y  * W0 + xbase + px]
          * inorm[(size_t)bt2 * HW0 + (size_t)yy * W0 + q];
      }
      dm[((size_t)pb * NDISP + dyi * 9 + dxi) * HW0 + (size_t)y * W0 + xbase + px] = v;
    }
    __syncthreads();
  }
}

// ---------------------------------------------------------------------------
// 3) separable 9x9 box aggregation (zero padding), H then V pass
// ---------------------------------------------------------------------------
__global__ void hbox_kernel(const float* __restrict__ in, float* __restrict__ out) {
  const long idx = (long)blockIdx.x * blockDim.x + threadIdx.x;
  if (idx >= (long)NPAIR * NDISP * HW0) return;
  const int p = (int)(idx % HW0);
  const int x = p % W0;
  const float* base = in + idx;
  float s = 0.f;
  #pragma unroll
  for (int d = -4; d <= 4; ++d) { const int xx = x + d; if (xx >= 0 && xx < W0) s += base[d]; }
  out[idx] = s;
}

__global__ void vbox_kernel(const float* __restrict__ in, float* __restrict__ out) {
  const long idx = (long)blockIdx.x * blockDim.x + threadIdx.x;
  if (idx >= (long)NPAIR * NDISP * HW0) return;
  const int p = (int)(idx % HW0);
  const int y = p / W0;
  const float* base = in + idx;
  float s = 0.f;
  #pragma unroll
  for (int d = -4; d <= 4; ++d) { const int yy = y + d; if (yy >= 0 && yy < H0) s += base[d * W0]; }
  out[idx] = s;
}

// ---------------------------------------------------------------------------
// 4) softmax over 81 displacements, expectation against directions,
//    divide by clip-id delta, mean over the 3 pairs -> flow (B,2,H0,W0)
// ---------------------------------------------------------------------------
__global__ void flow_kernel(const float* __restrict__ corr,
                            const float* __restrict__ past,
                            float* __restrict__ flow) {
  const int p = blockIdx.x * blockDim.x + threadIdx.x;
  const int b = blockIdx.y;
  if (p >= HW0) return;
  float accy = 0.f, accx = 0.f;
  for (int t = 0; t < TP - 1; ++t) {
    const float* base = corr + ((size_t)(b * (TP - 1) + t) * NDISP) * HW0 + p;
    float mx = -3.4e38f;
    for (int d = 0; d < NDISP; ++d) mx = fmaxf(mx, base[(size_t)d * HW0]);
    float sum = 0.f, gy = 0.f, gx = 0.f;
    for (int d = 0; d < NDISP; ++d) {
      const float e = expf(base[(size_t)d * HW0] - mx);
      sum += e;
      gy += e * (float)(d / 9 - 4);
      gx += e * (float)(d % 9 - 4);
    }
    const float delta = past[b * TP + t + 1] - past[b * TP + t];
    const float inv = 1.f / (sum * delta);
    accy += gy * inv; accx += gx * inv;
  }
  flow[((size_t)b * 2 + 0) * HW0 + p] = accy / (float)(TP - 1);
  flow[((size_t)b * 2 + 1) * HW0 + p] = accx / (float)(TP - 1);
}

// ---------------------------------------------------------------------------
// 5) bilinear resize (align_corners=True) of flow to a coarser level
// ---------------------------------------------------------------------------
__global__ void resize_kernel(const float* __restrict__ fin, float* __restrict__ fout,
                              int oh, int ow) {
  const int idx = blockIdx.x * blockDim.x + threadIdx.x;
  const int total = B_ * 2 * oh * ow;
  if (idx >= total) return;
  const int xo = idx % ow;
  const int yo = (idx / ow) % oh;
  const int bc = idx / (ow * oh);
  const float fy = yo * (float)(H0 - 1) / (float)(oh - 1);
  const float fx = xo * (float)(W0 - 1) / (float)(ow - 1);
  const int y0 = (int)floorf(fy), x0 = (int)floorf(fx);
  const int y1 = min(y0 + 1, H0 - 1), x1 = min(x0 + 1, W0 - 1);
  const float wy = fy - (float)y0, wx = fx - (float)x0;
  const float* src = fin + (size_t)bc * HW0;
  const float tl = src[y0 * W0 + x0], tr = src[y0 * W0 + x1];
  const float bl = src[y1 * W0 + x0], br = src[y1 * W0 + x1];
  fout[idx] = (tl * (1.f - wx) + tr * wx) * (1.f - wy) + (bl * (1.f - wx) + br * wx) * wy;
}

// ---------------------------------------------------------------------------
// 6) 1x1 conv (+BN fold +SiLU) as fp32 WMMA GEMM.
//    Block = (32,4): 4 waves, each owns a 16(Cout) x 16(pixel) tile.
//    Weights fetched as 8B-aligned v2f (global_load_b64); epilogue WMMA folds
//    the 2 flow channels with branch-free selects.
// ---------------------------------------------------------------------------
__global__ void __launch_bounds__(128)
conv1x1_wmma_kernel(const float* __restrict__ w,
                    const float* __restrict__ sc,
                    const float* __restrict__ bi,
                    const float* __restrict__ feat,
                    const float* __restrict__ flowl,
                    const float* __restrict__ fut,
                    float* __restrict__ out,
                    int C, int HW) {
  const int lane = threadIdx.x;
  const int half = lane >> 4;
  const int m    = lane & 15;
  const int mtile = blockIdx.y * blockDim.y + threadIdx.y;
  const int p = blockIdx.x * 16 + m;
  const int n = blockIdx.z;               // (b,t) flattened
  const int b = n / TF, t = n % TF;
  const int K = C + 2;                    // even, so &wrow[2*half*...] stays 8B aligned
  const float* fLast = feat + (size_t)(b * TP + (TP - 1)) * C * HW;
  const float* wrow  = w + (size_t)(mtile * 16 + m) * K;

  v8f acc = {0.f,0.f,0.f,0.f,0.f,0.f,0.f,0.f};
  const float* pa = wrow + 2 * half;
  const float* pb = fLast + (size_t)(2 * half) * HW + p;
  for (int k0 = 0; k0 < C; k0 += 4) {
    const v2f a = *(const v2f*)pa;        // one b64 load: w[o][ka], w[o][ka+1]
    v2f bv; bv.x = pb[0]; bv.y = pb[HW];
    acc = __builtin_amdgcn_wmma_f32_16x16x4_f32(false, a, false, bv, (short)0, acc, false, false);
    pa += 4;
    pb += (size_t)4 * HW;
  }
  { // epilogue: flow channels C, C+1 (zero pad K to 4), branch-free
    const float futv = fut[b * TF + t];
    const bool lo = (half == 0);
    const v2f aw = *(const v2f*)(wrow + C);
    const float c0 = flowl[((size_t)b * 2 + 0) * HW + p] * futv;
    const float c1 = flowl[((size_t)b * 2 + 1) * HW + p] * futv;
    v2f a, bv;
    a.x  = lo ? aw.x : 0.f;
    a.y  = lo ? aw.y : 0.f;
    bv.x = lo ? c0 : 0.f;
    bv.y = lo ? c1 : 0.f;
    acc = __builtin_amdgcn_wmma_f32_16x16x4_f32(false, a, false, bv, (short)0, acc, false, false);
  }
  #pragma unroll
  for (int r = 0; r < 8; ++r) {
    const int o = mtile * 16 + r + 8 * half;
    const float y = acc[r] * sc[o] + bi[o];
    out[((size_t)n * C + o) * HW + p] = y / (1.f + expf(-y));   // SiLU
  }
}

// ---------------------------------------------------------------------------
extern "C" void kernel_launch(void* const* d_in, const int* in_sizes, int n_in,
                              void* d_out, int out_size, void* d_ws, size_t ws_size,
                              hipStream_t stream) {
  (void)in_sizes; (void)n_in; (void)out_size; (void)ws_size;
  const float* f0   = (const float*)d_in[0];
  const float* w0   = (const float*)d_in[1];
  const float* s0   = (const float*)d_in[2];
  const float* b0   = (const float*)d_in[3];
  const float* f1   = (const float*)d_in[4];
  const float* w1   = (const float*)d_in[5];
  const float* s1   = (const float*)d_in[6];
  const float* b1   = (const float*)d_in[7];
  const float* f2   = (const float*)d_in[8];
  const float* w2   = (const float*)d_in[9];
  const float* s2   = (const float*)d_in[10];
  const float* b2   = (const float*)d_in[11];
  const float* past = (const float*)d_in[12];
  const float* fut  = (const float*)d_in[13];

  // workspace layout (floats): ~60.5 MB total
  float* ws    = (float*)d_ws;
  float* inorm = ws;                                   // 8 * HW0          = 122880
  float* bufA  = inorm + (size_t)B_ * TP * HW0;        // 6*81*HW0         = 7464960
  float* bufB  = bufA + (size_t)NPAIR * NDISP * HW0;   // 6*81*HW0         = 7464960
  float* flow0 = bufB + (size_t)NPAIR * NDISP * HW0;   // B*2*HW0          = 61440
  float* flow1 = flow0 + (size_t)B_ * 2 * HW0;         // B*2*HW1          = 15360
  float* flow2 = flow1 + (size_t)B_ * 2 * HW1;         // B*2*HW2          = 3840

  float* out0 = (float*)d_out;
  float* out1 = out0 + (size_t)B_ * TF * C0 * HW0;
  float* out2 = out1 + (size_t)B_ * TF * C1 * HW1;

  // 1) per-pixel inverse channel norms
  norm_kernel<<<dim3((HW0 + 255) / 256, B_ * TP), 256, 0, stream>>>(f0, inorm);

  // 2) WMMA correlation dot maps
  corr_dot_wmma_kernel<<<dim3(W0 / 16, H0, NPAIR), 32, 0, stream>>>(f0, inorm, bufA);

  // 3) separable 9x9 box aggregation
  const long nel = (long)NPAIR * NDISP * HW0;
  const int boxBlocks = (int)((nel + 255) / 256);
  hbox_kernel<<<boxBlocks, 256, 0, stream>>>(bufA, bufB);
  vbox_kernel<<<boxBlocks, 256, 0, stream>>>(bufB, bufA);

  // 4) softmax + expected displacement + pair mean -> flow (B,2,H0,W0)
  flow_kernel<<<dim3((HW0 + 255) / 256, B_), 256, 0, stream>>>(bufA, past, flow0);

  // 5) bilinear resizes for levels 1 and 2 (level 0 uses flow0 directly)
  resize_kernel<<<(B_ * 2 * HW1 + 255) / 256, 256, 0, stream>>>(flow0, flow1, H1, W1);
  resize_kernel<<<(B_ * 2 * HW2 + 255) / 256, 256, 0, stream>>>(flow0, flow2, H2, W2);

  // 6) fused concat + 1x1 conv + BN + SiLU (WMMA GEMM) per level
  conv1x1_wmma_kernel<<<dim3(HW0 / 16, C0 / 64, B_ * TF), dim3(32, 4), 0, stream>>>(
      w0, s0, b0, f0, flow0, fut, out0, C0, HW0);
  conv1x1_wmma_kernel<<<dim3(HW1 / 16, C1 / 64, B_ * TF), dim3(32, 4), 0, stream>>>(
      w1, s1, b1, f1, flow1, fut, out1, C1, HW1);
  conv1x1_wmma_kernel<<<dim3(HW2 / 16, C2 / 64, B_ * TF), dim3(32, 4), 0, stream>>>(
      w2, s2, b2, f2, flow2, fut, out2, C2, HW2);
}